// One_Track_GATModel_14087492731037
// MI455X (gfx1250) — compile-verified
//
#include <hip/hip_runtime.h>
#include <hip/hip_bf16.h>
#include <math.h>

typedef float v2f __attribute__((ext_vector_type(2)));
typedef float v8f __attribute__((ext_vector_type(8)));

#define WAVES_PER_BLOCK 8

// ---------------------------------------------------------------------------
// C[n,64] = A[n,K] @ B[K,64], fp32, via V_WMMA_F32_16X16X4_F32.
// One wave -> 16 rows x 64 cols. n must be a multiple of 16 (50000 = 3125*16).
// B is staged in LDS pre-swizzled: Bsw[n*(K+4)+k] = B[k][n], so each lane's
// {B[k][n],B[k+1][n]} pair is one conflict-free ds_load_b64.
// ---------------------------------------------------------------------------
__global__ void gemm_wmma(const float* __restrict__ A, const float* __restrict__ B,
                          float* __restrict__ C, int n, int K) {
    extern __shared__ float Bsw[];  // 64*(K+4) floats
    const int tid = threadIdx.x;
    const int ldb = K + 4;
    for (int i = tid; i < 64 * K; i += blockDim.x) {
        const int nn = i & 63;       // coalesced global read over n
        const int kk = i >> 6;
        Bsw[nn * ldb + kk] = B[(size_t)kk * 64 + nn];
    }
    __syncthreads();

    const int wave = tid >> 5;
    const int lane = tid & 31;
    const int row0 = (blockIdx.x * WAVES_PER_BLOCK + wave) * 16;
    if (row0 >= n) return;  // wave-uniform: EXEC stays all-ones for WMMA

    const int mrow  = lane & 15;        // M (for A) / N (for B,C)
    const int khalf = (lane >> 4) * 2;  // K sub-offset: 0 or 2

    v8f acc0 = {}, acc1 = {}, acc2 = {}, acc3 = {};
    const float* Arow = A + (size_t)(row0 + mrow) * K + khalf;
    const float* bp0  = Bsw + (mrow +  0) * ldb + khalf;
    const float* bp1  = Bsw + (mrow + 16) * ldb + khalf;
    const float* bp2  = Bsw + (mrow + 32) * ldb + khalf;
    const float* bp3  = Bsw + (mrow + 48) * ldb + khalf;

    for (int k0 = 0; k0 < K; k0 += 4) {
        const v2f a = *(const v2f*)(Arow + k0);
        v2f b;
        b = *(const v2f*)(bp0 + k0);
        acc0 = __builtin_amdgcn_wmma_f32_16x16x4_f32(false, a, false, b, (short)0, acc0, false, false);
        b = *(const v2f*)(bp1 + k0);
        acc1 = __builtin_amdgcn_wmma_f32_16x16x4_f32(false, a, false, b, (short)0, acc1, false, false);
        b = *(const v2f*)(bp2 + k0);
        acc2 = __builtin_amdgcn_wmma_f32_16x16x4_f32(false, a, false, b, (short)0, acc2, false, false);
        b = *(const v2f*)(bp3 + k0);
        acc3 = __builtin_amdgcn_wmma_f32_16x16x4_f32(false, a, false, b, (short)0, acc3, false, false);
    }

    // D layout: VGPR i -> row (i + 8*(lane/16)), col lane%16
    const int srow = (lane >> 4) * 8;
    const int col  = lane & 15;
#pragma unroll
    for (int i = 0; i < 8; ++i) {
        float* crow = C + (size_t)(row0 + srow + i) * 64 + col;
        crow[0]  = acc0[i];
        crow[16] = acc1[i];
        crow[32] = acc2[i];
        crow[48] = acc3[i];
    }
}

// ---------------------------------------------------------------------------
// Column sum + sum-of-squares over [n,64] into stats[0..63]=sum, [64..127]=sumsq
// ---------------------------------------------------------------------------
__global__ void col_stats(const float* __restrict__ Y, float* __restrict__ stats, int n) {
    const int col = threadIdx.x & 63;
    const int rpb = blockDim.x >> 6;
    int r = blockIdx.x * rpb + (threadIdx.x >> 6);
    const int stride = gridDim.x * rpb;
    float s = 0.f, ss = 0.f;
    for (; r < n; r += stride) {
        float v = Y[(size_t)r * 64 + col];
        s += v; ss += v * v;
    }
    atomicAdd(&stats[col], s);
    atomicAdd(&stats[64 + col], ss);
}

// ---------------------------------------------------------------------------
// Hout = elu( g*(Y-mean)*rsqrt(var+eps) + b ), biased var, ELU alpha=1
// ---------------------------------------------------------------------------
__global__ void bn_elu(const float* __restrict__ Y, float* __restrict__ Hout,
                       const float* __restrict__ stats, const float* __restrict__ g,
                       const float* __restrict__ bta, int n) {
    size_t i = (size_t)blockIdx.x * blockDim.x + threadIdx.x;
    if (i >= (size_t)n * 64) return;
    const int c = (int)(i & 63);
    const float inv_n = 1.0f / (float)n;
    const float m   = stats[c] * inv_n;
    const float var = stats[64 + c] * inv_n - m * m;
    const float z = g[c] * (Y[i] - m) * rsqrtf(var + 1e-5f) + bta[c];
    Hout[i] = z > 0.f ? z : expm1f(z);
}

// ---------------------------------------------------------------------------
// Per-layer init: acc = h (residual base), mx = -inf, den = 0, stats = 0
// ---------------------------------------------------------------------------
__global__ void init_layer(float* __restrict__ acc, const float* __restrict__ h,
                           float* __restrict__ mx, float* __restrict__ den,
                           float* __restrict__ stats, int n) {
    size_t i = (size_t)blockIdx.x * blockDim.x + threadIdx.x;
    if (i < (size_t)n * 64) acc[i] = h[i];
    if (i < (size_t)n * 8) { mx[i] = -3.0e38f; den[i] = 0.f; }
    if (i < 128) stats[i] = 0.f;
}

__global__ void zero_stats(float* __restrict__ stats) {
    if (threadIdx.x < 128) stats[threadIdx.x] = 0.f;
}

// ---------------------------------------------------------------------------
// Pass 1: logits[e,h] = sum_c leakyrelu(xl[src,h,c] + xr[dst,h,c]) * att[h,c]
//         segment max into mx[dst,h]
// ---------------------------------------------------------------------------
__global__ void edge_logits(const float* __restrict__ xl, const float* __restrict__ xr,
                            const int* __restrict__ src, const int* __restrict__ dst,
                            const float* __restrict__ att, float* __restrict__ exbuf,
                            float* __restrict__ mx, int E) {
    int t = blockIdx.x * blockDim.x + threadIdx.x;
    if (t >= E * 8) return;
    const int e  = t >> 3;
    const int hh = t & 7;
    const int s = src[e], d = dst[e];
    const float* pl = xl + (size_t)s * 64 + hh * 8;
    const float* pr = xr + (size_t)d * 64 + hh * 8;
    const float* pa = att + hh * 8;
    float logit = 0.f;
#pragma unroll
    for (int c = 0; c < 8; ++c) {
        float v = pl[c] + pr[c];
        v = v > 0.f ? v : 0.2f * v;
        logit += v * pa[c];
    }
    exbuf[t] = logit;
    atomicMax(&mx[(size_t)d * 8 + hh], logit);
}

// Pass 2: ex = exp(logit - mx[dst]); den[dst] += ex
__global__ void edge_exp(float* __restrict__ exbuf, const int* __restrict__ dst,
                         const float* __restrict__ mx, float* __restrict__ den, int E) {
    int t = blockIdx.x * blockDim.x + threadIdx.x;
    if (t >= E * 8) return;
    const int e  = t >> 3;
    const int hh = t & 7;
    const int d = dst[e];
    const float ex = __expf(exbuf[t] - mx[(size_t)d * 8 + hh]);
    exbuf[t] = ex;
    atomicAdd(&den[(size_t)d * 8 + hh], ex);
}

// Pass 3: alpha = ex/(den+1e-16); acc[dst, h*8+c] += xl[src, h*8+c] * alpha
__global__ void edge_scatter(const float* __restrict__ xl, const int* __restrict__ src,
                             const int* __restrict__ dst, const float* __restrict__ exbuf,
                             const float* __restrict__ den, float* __restrict__ acc, int E) {
    int t = blockIdx.x * blockDim.x + threadIdx.x;
    if (t >= E * 8) return;
    const int e  = t >> 3;
    const int hh = t & 7;
    const int s = src[e], d = dst[e];
    const float alpha = exbuf[t] / (den[(size_t)d * 8 + hh] + 1e-16f);
    const float* pl = xl + (size_t)s * 64 + hh * 8;
    float* po = acc + (size_t)d * 64 + hh * 8;
#pragma unroll
    for (int c = 0; c < 8; ++c) atomicAdd(&po[c], pl[c] * alpha);
}

// ---------------------------------------------------------------------------
// MLP head: 64 -> 48 -> 32 -> 16 -> 2 (elu on first 3), weights in LDS
// ---------------------------------------------------------------------------
__global__ void mlp_head(const float* __restrict__ H,
                         const float* __restrict__ W1, const float* __restrict__ B1,
                         const float* __restrict__ W2, const float* __restrict__ B2,
                         const float* __restrict__ W3, const float* __restrict__ B3,
                         const float* __restrict__ W4, const float* __restrict__ B4,
                         float* __restrict__ out, int n) {
    __shared__ float w1[64 * 48], w2[48 * 32], w3[32 * 16], w4[16 * 2];
    __shared__ float bb1[48], bb2[32], bb3[16], bb4[2];
    const int tid = threadIdx.x;
    for (int i = tid; i < 64 * 48; i += blockDim.x) w1[i] = W1[i];
    for (int i = tid; i < 48 * 32; i += blockDim.x) w2[i] = W2[i];
    for (int i = tid; i < 32 * 16; i += blockDim.x) w3[i] = W3[i];
    for (int i = tid; i < 16 * 2;  i += blockDim.x) w4[i] = W4[i];
    if (tid < 48) bb1[tid] = B1[tid];
    if (tid < 32) bb2[tid] = B2[tid];
    if (tid < 16) bb3[tid] = B3[tid];
    if (tid < 2)  bb4[tid] = B4[tid];
    __syncthreads();

    const int node = blockIdx.x * blockDim.x + tid;
    if (node >= n) return;

    float h0[64], t1[48], t2[32], t3[16];
    const float* hrow = H + (size_t)node * 64;
#pragma unroll
    for (int i = 0; i < 64; ++i) h0[i] = hrow[i];

    for (int j = 0; j < 48; ++j) {
        float s = bb1[j];
        for (int i = 0; i < 64; ++i) s += h0[i] * w1[i * 48 + j];
        t1[j] = s > 0.f ? s : expm1f(s);
    }
    for (int j = 0; j < 32; ++j) {
        float s = bb2[j];
        for (int i = 0; i < 48; ++i) s += t1[i] * w2[i * 32 + j];
        t2[j] = s > 0.f ? s : expm1f(s);
    }
    for (int j = 0; j < 16; ++j) {
        float s = bb3[j];
        for (int i = 0; i < 32; ++i) s += t2[i] * w3[i * 16 + j];
        t3[j] = s > 0.f ? s : expm1f(s);
    }
#pragma unroll
    for (int j = 0; j < 2; ++j) {
        float s = bb4[j];
        for (int i = 0; i < 16; ++i) s += t3[i] * w4[i * 2 + j];
        out[(size_t)node * 2 + j] = s;
    }
}

// ---------------------------------------------------------------------------
extern "C" void kernel_launch(void* const* d_in, const int* in_sizes, int n_in,
                              void* d_out, int out_size, void* d_ws, size_t ws_size,
                              hipStream_t stream) {
    const int N = in_sizes[0] / 128;
    const int E = in_sizes[1] / 2;

    const float* x    = (const float*)d_in[0];
    const int*   eidx = (const int*)d_in[1];
    const int*   src  = eidx;
    const int*   dst  = eidx + E;
    const float* Wpre = (const float*)d_in[2];
    const float* g0   = (const float*)d_in[3];
    const float* b0   = (const float*)d_in[4];
    const float* Wp1 = (const float*)d_in[25]; const float* bp1 = (const float*)d_in[26];
    const float* Wp2 = (const float*)d_in[27]; const float* bp2 = (const float*)d_in[28];
    const float* Wp3 = (const float*)d_in[29]; const float* bp3 = (const float*)d_in[30];
    const float* Wp4 = (const float*)d_in[31]; const float* bp4 = (const float*)d_in[32];
    float* out = (float*)d_out;

    // Workspace layout (floats)
    const size_t N64 = (size_t)N * 64;
    const size_t N8  = (size_t)N * 8;
    float* ws    = (float*)d_ws;
    float* h     = ws;              // N*64
    float* acc   = h + N64;         // N*64
    float* xl    = acc + N64;       // N*64
    float* xr    = xl + N64;        // N*64
    float* mx    = xr + N64;        // N*8
    float* den   = mx + N8;         // N*8
    float* stats = den + N8;        // 128
    float* exbuf = stats + 128;     // E*8

    const int gemmBlocks = ((N / 16) + WAVES_PER_BLOCK - 1) / WAVES_PER_BLOCK;
    const int elemBlocks = (int)((N64 + 255) / 256);
    const int edgeBlocks = (E * 8 + 255) / 256;
    const int statBlocks = 256;

    const size_t ldsPre = 64 * (128 + 4) * sizeof(float);
    const size_t ldsGat = 64 * (64 + 4) * sizeof(float);

    // ---- pre layer: acc = x @ Wpre; h = elu(bn(acc)) ----
    gemm_wmma<<<gemmBlocks, 256, ldsPre, stream>>>(x, Wpre, acc, N, 128);
    zero_stats<<<1, 128, 0, stream>>>(stats);
    col_stats<<<statBlocks, 256, 0, stream>>>(acc, stats, N);
    bn_elu<<<elemBlocks, 256, 0, stream>>>(acc, h, stats, g0, b0, N);

    // ---- 4 GATv2 layers ----
    for (int L = 0; L < 4; ++L) {
        const float* Wl  = (const float*)d_in[5 + 5 * L + 0];
        const float* Wr  = (const float*)d_in[5 + 5 * L + 1];
        const float* att = (const float*)d_in[5 + 5 * L + 2];
        const float* g   = (const float*)d_in[5 + 5 * L + 3];
        const float* b   = (const float*)d_in[5 + 5 * L + 4];

        gemm_wmma<<<gemmBlocks, 256, ldsGat, stream>>>(h, Wl, xl, N, 64);
        gemm_wmma<<<gemmBlocks, 256, ldsGat, stream>>>(h, Wr, xr, N, 64);
        init_layer<<<elemBlocks, 256, 0, stream>>>(acc, h, mx, den, stats, N);

        edge_logits<<<edgeBlocks, 256, 0, stream>>>(xl, xr, src, dst, att, exbuf, mx, E);
        edge_exp<<<edgeBlocks, 256, 0, stream>>>(exbuf, dst, mx, den, E);
        edge_scatter<<<edgeBlocks, 256, 0, stream>>>(xl, src, dst, exbuf, den, acc, E);

        col_stats<<<statBlocks, 256, 0, stream>>>(acc, stats, N);
        bn_elu<<<elemBlocks, 256, 0, stream>>>(acc, h, stats, g, b, N);
    }

    // ---- MLP head ----
    mlp_head<<<(N + 255) / 256, 256, 0, stream>>>(h, Wp1, bp1, Wp2, bp2, Wp3, bp3, Wp4, bp4, out, N);
}